// Attention_45956150067876
// MI455X (gfx1250) — compile-verified
//
#include <hip/hip_runtime.h>
#include <hip/hip_bf16.h>
#include <math.h>

// ---------------------------------------------------------------------------
// Decode attention: B=16, S=4096, H=4096, NH=32, NKV=8, HD=128, NREP=4
// FP32 end-to-end. Bandwidth-bound: ~700MB mandatory traffic @ 23.3 TB/s
// => ~30us ceiling; WMMA f32 16x16x4 used for both skinny GEMMs (M=16=batch).
// Split-K x4 on the GEMMs for memory-level parallelism; non-temporal loads
// for all stream-once data so the 192MB L2 keeps the reused small buffers.
// ---------------------------------------------------------------------------

#define B_   16
#define S_   4096
#define H_   4096
#define NH_  32
#define NKV_ 8
#define HD_  128
#define NREP_ 4
#define QKV_O ((NH_ + 2 * NKV_) * HD_)   // 6144
#define Q_END (NH_ * HD_)                // 4096
#define K_END (Q_END + NKV_ * HD_)       // 5120
#define KSPLIT 4

typedef __attribute__((ext_vector_type(2))) float v2f;
typedef __attribute__((ext_vector_type(4))) float v4f;
typedef __attribute__((ext_vector_type(8))) float v8f;

// ---------------------------------------------------------------------------
// Kernel: partial C[split][16 x N] = A[16 x Kslice] * W^T, W row-major [N x K].
// One wave per (16-wide N tile, K slice). WMMA f32 16x16x4.
// Fragment layout (ISA 7.12.2, 32-bit):
//   A 16x4 : lane L holds row M=L%16; VGPR0=K(2*(L/16)), VGPR1=K(2*(L/16)+1)
//   B 4x16 : lane L holds col N=L%16; same K striping
//   C 16x16: VGPR r -> M = r + 8*(L/16), N = L%16
// W fragments are loaded non-temporal (streamed exactly once).
// ---------------------------------------------------------------------------
__global__ void gemm16_wmma_split(const float* __restrict__ A,
                                  const float* __restrict__ W,
                                  float* __restrict__ P,   // [KSPLIT][16][ldc]
                                  int K, int ldc) {
    const int lane  = threadIdx.x;
    const int nl    = lane & 15;
    const int hi    = lane >> 4;          // 0 or 1
    const int nbase = blockIdx.x * 16;
    const int Kc    = K / KSPLIT;
    const int k0    = blockIdx.y * Kc;

    const float* ap = A + (size_t)nl * K + k0 + 2 * hi;
    const float* wp = W + (size_t)(nbase + nl) * K + k0 + 2 * hi;

    v8f acc = {0.f, 0.f, 0.f, 0.f, 0.f, 0.f, 0.f, 0.f};

    for (int kb = 0; kb < Kc; kb += 16) {
#pragma unroll
        for (int u = 0; u < 4; ++u) {
            const int kk = kb + 4 * u;
            v2f a = *(const v2f*)(ap + kk);
            v2f b = __builtin_nontemporal_load((const v2f*)(wp + kk));
            // (neg_a, A, neg_b, B, c_mod, C, reuse_a, reuse_b)
            acc = __builtin_amdgcn_wmma_f32_16x16x4_f32(
                false, a, false, b, (short)0, acc, false, false);
        }
    }

#pragma unroll
    for (int r = 0; r < 8; ++r) {
        P[((size_t)blockIdx.y * 16 + r + 8 * hi) * ldc + nbase + nl] = acc[r];
    }
}

// Deterministic split-K reduction: C[i] = sum_s P[s][i] (fixed order).
__global__ void reduce_split(const float* __restrict__ P,
                             float* __restrict__ C, int n4) {
    const int i = blockIdx.x * blockDim.x + threadIdx.x;
    if (i >= n4) return;
    const v4f* p = (const v4f*)P;
    v4f s = p[i];
#pragma unroll
    for (int w = 1; w < KSPLIT; ++w) s += p[(size_t)w * n4 + i];
    ((v4f*)C)[i] = s;
}

// ---------------------------------------------------------------------------
// RoPE + RMSNorm for q and k heads. One wave per head; lane owns 4 elements
// (= 2 rope pairs). tasks: [0,32) -> q head, [32,40) -> k head.
// ---------------------------------------------------------------------------
__global__ void rope_rms(const float* __restrict__ qkv,
                         const int* __restrict__ last_pos,
                         const float* __restrict__ rope_cache,
                         float* __restrict__ qn,
                         float* __restrict__ knew) {
    const int task = blockIdx.x;
    const int b = task / (NH_ + NKV_);
    const int h = task % (NH_ + NKV_);
    const int pos = last_pos[b];
    const int lane = threadIdx.x;

    const float* src;
    float* dst;
    if (h < NH_) {
        src = qkv + (size_t)b * QKV_O + h * HD_;
        dst = qn + ((size_t)b * NH_ + h) * HD_;
    } else {
        const int kv = h - NH_;
        src = qkv + (size_t)b * QKV_O + Q_END + kv * HD_;
        dst = knew + ((size_t)b * NKV_ + kv) * HD_;
    }

    const v4f xv = *(const v4f*)(src + 4 * lane);
    // rope_cache[pos, j, 0/1] = (cos, sin); 4 floats = pairs j=2L, 2L+1
    const v4f rc = *(const v4f*)(rope_cache + (size_t)pos * HD_ + 4 * lane);

    const float r0 = xv[0] * rc[0] - xv[1] * rc[1];
    const float i0 = xv[1] * rc[0] + xv[0] * rc[1];
    const float r1 = xv[2] * rc[2] - xv[3] * rc[3];
    const float i1 = xv[3] * rc[2] + xv[2] * rc[3];

    float ss = r0 * r0 + i0 * i0 + r1 * r1 + i1 * i1;
#pragma unroll
    for (int off = 16; off > 0; off >>= 1)
        ss += __shfl_xor(ss, off, 32);

    const float scale = rsqrtf(ss * (1.0f / (float)HD_) + 1e-5f);

    v4f o;
    o[0] = r0 * scale; o[1] = i0 * scale; o[2] = r1 * scale; o[3] = i1 * scale;
    *(v4f*)(dst + 4 * lane) = o;
}

__device__ __forceinline__ float dot4(const v4f a, const v4f b) {
    return a[0] * b[0] + a[1] * b[1] + a[2] * b[2] + a[3] * b[3];
}

// ---------------------------------------------------------------------------
// GQA flash decode. One block per (b, kv-group); 8 waves, each owning 512
// positions with an online softmax; merge via LDS. Lane owns d=4*lane..+3 of
// the 128-dim head -> coalesced 512B/wave K/V reads, non-temporal (the 536MB
// cache stream is touched once). Row at last_pos substituted with fresh k/v.
// ---------------------------------------------------------------------------
__global__ void attn_decode(const float* __restrict__ qn,
                            const float* __restrict__ knew,
                            const float* __restrict__ qkv,      // fresh v lives here
                            const float* __restrict__ cache_k,
                            const float* __restrict__ cache_v,
                            const int* __restrict__ last_pos,
                            const unsigned char* __restrict__ mask,
                            float* __restrict__ aout) {
    const int b = blockIdx.x / NKV_;
    const int g = blockIdx.x % NKV_;
    const int wave = threadIdx.x >> 5;
    const int lane = threadIdx.x & 31;
    const int pos = last_pos[b];

    v4f q[NREP_];
#pragma unroll
    for (int r = 0; r < NREP_; ++r)
        q[r] = *(const v4f*)(qn + ((size_t)(b * NH_ + g * NREP_ + r)) * HD_ + 4 * lane);

    const size_t srow = (size_t)NKV_ * HD_;  // 1024 floats between s positions
    const float* kb0 = cache_k + ((size_t)b * S_) * srow + (size_t)g * HD_ + 4 * lane;
    const float* vb0 = cache_v + ((size_t)b * S_) * srow + (size_t)g * HD_ + 4 * lane;
    const float* knp = knew + ((size_t)b * NKV_ + g) * HD_ + 4 * lane;
    const float* vnp = qkv + (size_t)b * QKV_O + K_END + g * HD_ + 4 * lane;
    const unsigned char* mrow = mask + (size_t)b * S_;

    const float sc = 0.08838834764831845f;   // 1/sqrt(128)

    float m = -INFINITY, l = 0.f;
    v4f acc[NREP_];
#pragma unroll
    for (int r = 0; r < NREP_; ++r) acc[r] = (v4f){0.f, 0.f, 0.f, 0.f};

    const int s0 = wave * (S_ / 8);
    for (int s = s0; s < s0 + (S_ / 8); ++s) {
        if (!mrow[s]) continue;   // masked -> score -inf -> contributes 0
        const bool fresh = (s == pos);
        v4f k4, v4;
        if (fresh) {
            k4 = *(const v4f*)knp;
            v4 = *(const v4f*)vnp;
        } else {
            k4 = __builtin_nontemporal_load((const v4f*)(kb0 + (size_t)s * srow));
            v4 = __builtin_nontemporal_load((const v4f*)(vb0 + (size_t)s * srow));
        }

        float d0 = dot4(q[0], k4);
        float d1 = dot4(q[1], k4);
        float d2 = dot4(q[2], k4);
        float d3 = dot4(q[3], k4);
#pragma unroll
        for (int off = 16; off > 0; off >>= 1) {
            d0 += __shfl_xor(d0, off, 32);
            d1 += __shfl_xor(d1, off, 32);
            d2 += __shfl_xor(d2, off, 32);
            d3 += __shfl_xor(d3, off, 32);
        }
        d0 *= sc; d1 *= sc; d2 *= sc; d3 *= sc;

        const float mnew = fmaxf(m, fmaxf(fmaxf(d0, d1), fmaxf(d2, d3)));
        const float corr = __expf(m - mnew);     // 0 on first iteration
        const float p0 = __expf(d0 - mnew);
        const float p1 = __expf(d1 - mnew);
        const float p2 = __expf(d2 - mnew);
        const float p3 = __expf(d3 - mnew);
        l = l * corr + (p0 + p1 + p2 + p3);

        acc[0] = acc[0] * corr + p0 * v4;
        acc[1] = acc[1] * corr + p1 * v4;
        acc[2] = acc[2] * corr + p2 * v4;
        acc[3] = acc[3] * corr + p3 * v4;
        m = mnew;
    }

    // ---- cross-wave merge ----
    __shared__ float sm[8];
    __shared__ float sl[8];
    __shared__ float sacc[8][NREP_][HD_];

    if (lane == 0) { sm[wave] = m; sl[wave] = l; }
#pragma unroll
    for (int r = 0; r < NREP_; ++r)
        *(v4f*)&sacc[wave][r][4 * lane] = acc[r];
    __syncthreads();

    float M = sm[0];
#pragma unroll
    for (int w = 1; w < 8; ++w) M = fmaxf(M, sm[w]);
    float L = 0.f;
    float f[8];
#pragma unroll
    for (int w = 0; w < 8; ++w) { f[w] = __expf(sm[w] - M); L += sl[w] * f[w]; }
    const float invL = 1.0f / L;

    for (int e = threadIdx.x; e < NREP_ * HD_; e += blockDim.x) {
        const int r = e >> 7;
        const int d = e & (HD_ - 1);
        float sum = 0.f;
#pragma unroll
        for (int w = 0; w < 8; ++w) sum += sacc[w][r][d] * f[w];
        aout[(size_t)b * (NH_ * HD_) + (g * NREP_ + r) * HD_ + d] = sum * invL;
    }
}

// ---------------------------------------------------------------------------
// Launch: inputs (x, last_pos, rope_cache, mask, wqkv, o_proj_w,
//                 cache_k, cache_v); output fp32 [16, 4096].
// ---------------------------------------------------------------------------
extern "C" void kernel_launch(void* const* d_in, const int* in_sizes, int n_in,
                              void* d_out, int out_size, void* d_ws, size_t ws_size,
                              hipStream_t stream) {
    const float* x          = (const float*)d_in[0];
    const int*   last_pos   = (const int*)d_in[1];
    const float* rope_cache = (const float*)d_in[2];
    const unsigned char* mask = (const unsigned char*)d_in[3];
    const float* wqkv       = (const float*)d_in[4];
    const float* o_proj_w   = (const float*)d_in[5];
    const float* cache_k    = (const float*)d_in[6];
    const float* cache_v    = (const float*)d_in[7];
    float* out = (float*)d_out;

    float* ws   = (float*)d_ws;
    float* qkv  = ws;                             // 16*6144
    float* qn   = qkv + (size_t)B_ * QKV_O;       // 16*32*128
    float* knew = qn + (size_t)B_ * NH_ * HD_;    // 16*8*128
    float* aout = knew + (size_t)B_ * NKV_ * HD_; // 16*4096
    float* part = aout + (size_t)B_ * NH_ * HD_;  // KSPLIT*16*6144 (shared scratch)

    // 1) qkv = x @ wqkv^T   (N=6144, K=4096), split-K x4 + deterministic reduce
    gemm16_wmma_split<<<dim3(QKV_O / 16, KSPLIT), 32, 0, stream>>>(
        x, wqkv, part, H_, QKV_O);
    reduce_split<<<(B_ * QKV_O / 4 + 255) / 256, 256, 0, stream>>>(
        part, qkv, B_ * QKV_O / 4);

    // 2) rope + rmsnorm on q and k
    rope_rms<<<B_ * (NH_ + NKV_), 32, 0, stream>>>(qkv, last_pos, rope_cache, qn, knew);

    // 3) flash GQA decode over the cache (fresh row substituted in-flight)
    attn_decode<<<B_ * NKV_, 256, 0, stream>>>(qn, knew, qkv, cache_k, cache_v,
                                               last_pos, mask, aout);

    // 4) out = aout @ o_proj_w^T   (N=4096, K=4096), split-K x4 + reduce
    gemm16_wmma_split<<<dim3(H_ / 16, KSPLIT), 32, 0, stream>>>(
        aout, o_proj_w, part, H_, H_);
    reduce_split<<<(B_ * H_ / 4 + 255) / 256, 256, 0, stream>>>(
        part, out, B_ * H_ / 4);
}